// DecoupledGCNUnit_7413113553419
// MI455X (gfx1250) — compile-verified
//
#include <hip/hip_runtime.h>
#include <hip/hip_bf16.h>

// ---------------------------------------------------------------------------
// DecoupledGCN unit on gfx1250 (MI455X), f32 via V_WMMA_F32_16X16X4_F32.
//
// Shapes: x0 [N=32, C=64, T=288, V=25]  (P = T*V = 7200 positions per n)
//         W  [64, 192], bias[192], decoupled_A [3,8,25,25]
// Pipeline (stream-ordered): stats-zero -> WMMA GEMM(+bn0 stats) ->
//   bn0 finalize + adjacency normalize -> WMMA graph conv(+bn1 stats) ->
//   bn1 finalize -> fused residual+relu epilogue.
//
// Scheduling notes:
//  * WMMA operand loads are staged into distinct register arrays per k-phase
//    so they issue as load clauses with partial s_wait_loadcnt values.
//  * K1 uses 6 accumulator tiles per wave (96 channels) instead of 12 to stay
//    under the 128-VGPR occupancy cap -> no scratch spills in the hot loop.
//    The channel-group bit is the LSB of the wave index so the two waves
//    sharing identical x0 B-fragments are adjacent (L0-resident reuse).
// ---------------------------------------------------------------------------

typedef __attribute__((ext_vector_type(2))) float v2f;
typedef __attribute__((ext_vector_type(8))) float v8f;

#define NB     32
#define CIN    64
#define TT     288
#define VV     25
#define DCH    192        // OUT_CH * NUM_SUBSET
#define OCH    64
#define PPN    (TT*VV)    // 7200 positions per batch item
#define NPOS   (NB*PPN)   // 230400 positions total (BN population)

// workspace layout (float offsets)
#define X_OFF   ((size_t)0)                       // x  : 32*192*7200 = 44,236,800
#define Y_OFF   ((size_t)44236800)                // y  : 32*64*7200  = 14,745,600
#define NA_OFF  ((size_t)58982400)                // normA : 24*625 = 15,000
#define ST_OFF  ((size_t)58997400)                // xstats 384 | ystats 128  (zeroed)
#define P_OFF   ((size_t)58997912)                // scale0/shift0 384 | scale1/shift1 128

__device__ __forceinline__ v8f wmma4f32(v2f a, v2f b, v8f c) {
  return __builtin_amdgcn_wmma_f32_16x16x4_f32(false, a, false, b, (short)0, c,
                                               false, false);
}

// --- K0: zero the 512-float stats region ------------------------------------
__global__ void k0_zero(float* stats) {
  int i = threadIdx.x;
  if (i < 512) stats[i] = 0.0f;
}

// --- K1: x = x0*W + b, WMMA GEMM; fused per-channel sum/sumsq (bn0 stats) ---
// grid: 3600 blocks x 256 threads = 28800 waves; one wave = 16 positions,
// 96 output channels (6 M-tiles), K = 64 in 16 steps of 4.
__global__ void __launch_bounds__(256)
k1_gemm(const float* __restrict__ x0,
        const float* __restrict__ W,
        const float* __restrict__ bias,
        float* __restrict__ xbuf,
        float* __restrict__ xstats) {
  __shared__ float s[384];
  for (int i = threadIdx.x; i < 384; i += blockDim.x) s[i] = 0.0f;
  __syncthreads();

  const int lane = threadIdx.x & 31;
  const int l15  = lane & 15;
  const int half = lane >> 4;

  const int wave  = blockIdx.x * 8 + (threadIdx.x >> 5);   // 0..28799
  const int dgrp  = wave & 1;                              // channel half: 0/1
  const int rem   = wave >> 1;                             // 0..14399
  const int n     = rem / 450;
  const int p0    = (rem % 450) * 16;
  const int dbase = 96 * dgrp;

  const float* x0n = x0 + (size_t)n * CIN * PPN + p0 + l15;

  v8f acc[6];
#pragma unroll
  for (int j = 0; j < 6; ++j) acc[j] = (v8f){0,0,0,0,0,0,0,0};

#pragma unroll
  for (int ks = 0; ks < 16; ++ks) {                // K step: c0 = 4*ks + 2*half
    const int c0 = 4 * ks + 2 * half;
    // stage B (x0) and all 6 A (W) fragments into distinct registers first
    v2f b;
    b[0] = x0n[(size_t)c0 * PPN];
    b[1] = x0n[(size_t)(c0 + 1) * PPN];
    const float* w0 = W + (size_t)c0 * DCH + dbase + l15;  // row c0
    const float* w1 = w0 + DCH;                            // row c0+1
    v2f a[6];
#pragma unroll
    for (int j = 0; j < 6; ++j) {
      a[j][0] = w0[16 * j];
      a[j][1] = w1[16 * j];
    }
#pragma unroll
    for (int j = 0; j < 6; ++j) acc[j] = wmma4f32(a[j], b, acc[j]);
  }

  // epilogue: add bias, store x, shuffle-reduce per-channel sum/sumsq
#pragma unroll
  for (int j = 0; j < 6; ++j) {
#pragma unroll
    for (int r = 0; r < 8; ++r) {
      const int d = dbase + 16 * j + r + 8 * half;
      float val = acc[j][r] + bias[d];
      xbuf[((size_t)n * DCH + d) * PPN + p0 + l15] = val;
      float sm = val, sq = val * val;
#pragma unroll
      for (int m = 1; m < 16; m <<= 1) {           // stays within 16-lane half
        sm += __shfl_xor(sm, m, 32);
        sq += __shfl_xor(sq, m, 32);
      }
      if (l15 == 0) {
        atomicAdd(&s[d], sm);
        atomicAdd(&s[192 + d], sq);
      }
    }
  }
  __syncthreads();
  for (int i = threadIdx.x; i < 384; i += blockDim.x) atomicAdd(&xstats[i], s[i]);
}

// --- K2: finalize bn0 (scale/shift per d) + column-normalize adjacency -----
__global__ void k2_prep(const float* __restrict__ xstats,
                        const float* __restrict__ bn0w,
                        const float* __restrict__ bn0b,
                        const float* __restrict__ dA,
                        float* __restrict__ s0,      // scale0[192], shift0[192]
                        float* __restrict__ normA) { // [24][25][25]
  const int t = threadIdx.x;
  const float Np = (float)NPOS;
  if (t < DCH) {
    float mean = xstats[t] / Np;
    float var  = xstats[DCH + t] / Np - mean * mean;
    float sc   = bn0w[t] * rsqrtf(var + 1e-5f);
    s0[t]        = sc;
    s0[DCH + t]  = bn0b[t] - mean * sc;
  } else if (t < DCH + 600) {                      // 24 matrices x 25 columns
    const int col = t - DCH;
    const int w   = col % 25;
    const int kg  = col / 25;
    const float* A = dA + kg * 625;
    float dsum = 0.0f;
    for (int v = 0; v < 25; ++v) dsum += A[v * 25 + w];
    const float inv = 1.0f / (dsum + 0.001f);
    for (int v = 0; v < 25; ++v) normA[kg * 625 + v * 25 + w] = A[v * 25 + w] * inv;
  }
}

// --- K3: graph conv y[n,c,t,w] = sum_{k,v} bn0(x)[n,kc,t,v] * Ahat[k,c,v,w] -
// WMMA: M=16 (t tile), K=25 padded to 28 (7 steps), N=25 as 16+9 tiles.
// grid: 4608 blocks x 256 threads = 36864 waves; wave = one (n, c, t-tile).
__global__ void __launch_bounds__(256)
k3_conv(const float* __restrict__ xbuf,
        const float* __restrict__ normA,
        const float* __restrict__ s0,
        float* __restrict__ ybuf,
        float* __restrict__ ystats) {
  const int lane = threadIdx.x & 31;
  const int l15  = lane & 15;
  const int half = lane >> 4;

  const int wave  = blockIdx.x * 8 + (threadIdx.x >> 5);   // 0..36863
  const int ttile = wave % 18;
  const int c     = (wave / 18) % OCH;
  const int n     = wave / (18 * OCH);
  const int t0    = ttile * 16;
  const int g     = c & 7;                                 // tile over groups

  v8f acc0 = (v8f){0,0,0,0,0,0,0,0};
  v8f acc1 = (v8f){0,0,0,0,0,0,0,0};

#pragma unroll
  for (int kk = 0; kk < 3; ++kk) {
    const int d = kk * OCH + c;
    const float sc = s0[d];
    const float sh = s0[DCH + d];
    const float* xrow = xbuf + ((size_t)(n * DCH + d) * TT + t0 + l15) * VV;
    const float* Amat = normA + (kk * 8 + g) * 625;

    // stage all 7 k-steps of A and B fragments, then issue 14 WMMAs
    v2f a[7], b0[7], b1[7];
#pragma unroll
    for (int ks = 0; ks < 7; ++ks) {
      const int v0 = 4 * ks + 2 * half;                    // padded K: 0..27
      a[ks][0] = 0.0f; a[ks][1] = 0.0f;
      b0[ks][0] = 0.0f; b0[ks][1] = 0.0f;
      b1[ks][0] = 0.0f; b1[ks][1] = 0.0f;
      if (v0 < VV) {
        a[ks][0]  = xrow[v0] * sc + sh;
        b0[ks][0] = Amat[v0 * VV + l15];
        if (l15 + 16 < VV) b1[ks][0] = Amat[v0 * VV + l15 + 16];
      }
      if (v0 + 1 < VV) {
        a[ks][1]  = xrow[v0 + 1] * sc + sh;
        b0[ks][1] = Amat[(v0 + 1) * VV + l15];
        if (l15 + 16 < VV) b1[ks][1] = Amat[(v0 + 1) * VV + l15 + 16];
      }
    }
#pragma unroll
    for (int ks = 0; ks < 7; ++ks) {
      acc0 = wmma4f32(a[ks], b0[ks], acc0);
      acc1 = wmma4f32(a[ks], b1[ks], acc1);
    }
  }

  // epilogue: store y, accumulate bn1 stats for channel c
  float sm = 0.0f, sq = 0.0f;
  float* yc = ybuf + (size_t)(n * OCH + c) * PPN;
#pragma unroll
  for (int r = 0; r < 8; ++r) {
    const int t = t0 + r + 8 * half;
    const float v0 = acc0[r];
    yc[(size_t)t * VV + l15] = v0;                          // w = 0..15 (<25)
    sm += v0; sq += v0 * v0;
    if (l15 < 9) {                                          // w = 16..24
      const float v1 = acc1[r];
      yc[(size_t)t * VV + 16 + l15] = v1;
      sm += v1; sq += v1 * v1;
    }
  }
#pragma unroll
  for (int m = 1; m < 32; m <<= 1) {
    sm += __shfl_xor(sm, m, 32);
    sq += __shfl_xor(sq, m, 32);
  }
  if (lane == 0) {
    atomicAdd(&ystats[c], sm);
    atomicAdd(&ystats[OCH + c], sq);
  }
}

// --- K4: finalize bn1 -------------------------------------------------------
__global__ void k4_bn1(const float* __restrict__ ystats,
                       const float* __restrict__ bnw,
                       const float* __restrict__ bnb,
                       float* __restrict__ s1) {
  const int t = threadIdx.x;
  if (t < OCH) {
    const float Np = (float)NPOS;
    float mean = ystats[t] / Np;
    float var  = ystats[OCH + t] / Np - mean * mean;
    float sc   = bnw[t] * rsqrtf(var + 1e-5f);
    s1[t]       = sc;
    s1[OCH + t] = bnb[t] - mean * sc;
  }
}

// --- K5: out = relu(bn1(y) + x0), float4-vectorized -------------------------
__global__ void k5_epilogue(const float* __restrict__ ybuf,
                            const float* __restrict__ x0,
                            const float* __restrict__ s1,
                            float* __restrict__ out) {
  const size_t i4 = (size_t)blockIdx.x * blockDim.x + threadIdx.x; // 3,686,400
  const int c = (int)((i4 * 4) / PPN) % OCH;                       // PPN%4==0
  const float sc = s1[c], sh = s1[OCH + c];
  const float4 y = ((const float4*)ybuf)[i4];
  const float4 x = ((const float4*)x0)[i4];
  float4 r;
  r.x = fmaxf(y.x * sc + sh + x.x, 0.0f);
  r.y = fmaxf(y.y * sc + sh + x.y, 0.0f);
  r.z = fmaxf(y.z * sc + sh + x.z, 0.0f);
  r.w = fmaxf(y.w * sc + sh + x.w, 0.0f);
  ((float4*)out)[i4] = r;
}

extern "C" void kernel_launch(void* const* d_in, const int* in_sizes, int n_in,
                              void* d_out, int out_size, void* d_ws, size_t ws_size,
                              hipStream_t stream) {
  (void)in_sizes; (void)n_in; (void)out_size; (void)ws_size;
  const float* x0    = (const float*)d_in[0];
  const float* dA    = (const float*)d_in[1];
  const float* W     = (const float*)d_in[2];
  const float* bias  = (const float*)d_in[3];
  const float* bn0w  = (const float*)d_in[4];
  const float* bn0b  = (const float*)d_in[5];
  const float* bnw   = (const float*)d_in[6];
  const float* bnb   = (const float*)d_in[7];
  float* out = (float*)d_out;

  float* ws     = (float*)d_ws;
  float* xbuf   = ws + X_OFF;
  float* ybuf   = ws + Y_OFF;
  float* normA  = ws + NA_OFF;
  float* xstats = ws + ST_OFF;           // 384
  float* ystats = ws + ST_OFF + 384;     // 128
  float* s0     = ws + P_OFF;            // 384
  float* s1     = ws + P_OFF + 384;      // 128

  k0_zero<<<1, 512, 0, stream>>>(xstats);
  k1_gemm<<<3600, 256, 0, stream>>>(x0, W, bias, xbuf, xstats);
  k2_prep<<<1, 1024, 0, stream>>>(xstats, bn0w, bn0b, dA, s0, normA);
  k3_conv<<<4608, 256, 0, stream>>>(xbuf, normA, s0, ybuf, ystats);
  k4_bn1<<<1, 64, 0, stream>>>(ystats, bnw, bnb, s1);
  k5_epilogue<<<14400, 256, 0, stream>>>(ybuf, x0, s1, out);
}